// AutoregressiveAllocPolicy_2551210573967
// MI455X (gfx1250) — compile-verified
//
#include <hip/hip_runtime.h>
#include <stdint.h>
#include <math.h>

typedef float v2f __attribute__((ext_vector_type(2)));
typedef float v8f __attribute__((ext_vector_type(8)));
typedef unsigned int u32x4 __attribute__((ext_vector_type(4)));
typedef int i32x4 __attribute__((ext_vector_type(4)));
typedef int i32x8 __attribute__((ext_vector_type(8)));

#define NB 16   // batches per workgroup (= WMMA M)
#define NT 16   // tasks
#define NA 16   // agents (serial steps)
#define DD 128  // embed dim
#define K2 256  // 2*DD (GEMM K)

__global__ __launch_bounds__(32)
void alloc_policy_kernel(const float* __restrict__ task_embeds,
                         const float* __restrict__ nonag_g,
                         const float* __restrict__ agent_embeds,
                         const uint8_t* __restrict__ task_mask,
                         const float* __restrict__ agent_mask,
                         const float* __restrict__ gumbels,
                         const float* __restrict__ W_count,
                         const float* __restrict__ b_count,
                         const float* __restrict__ W_upd,
                         const float* __restrict__ b_upd,
                         float* __restrict__ out, int B)
{
    // LDS-resident state for the whole 16-step scan.
    __shared__ float te_s[NB * NT * DD];   // 128 KB evolving task embeddings
    __shared__ float nonag_s[NB * NT];
    __shared__ float counts_s[NB * NT];
    __shared__ float wc0_s[DD], wc1_s[DD], bc_s[DD], bu_s[DD];
    __shared__ int   ts_s[NB];             // per-batch argmax task
    __shared__ float f_s[NB];              // per-batch te-update factor
    __shared__ uint32_t mbits_s[NB];       // task_mask bits per batch
    // Declared LAST: TDM demo target + per-step A-matrix staging. Any TDM
    // overrun falls past the LDS allocation and is dropped by hardware.
    __shared__ float xT_s[K2 * NB];        // 16 KB A-matrix, transposed [k][m]

    const int lane = threadIdx.x;          // 0..31 (wave32)
    const int b    = lane & 15;            // batch-in-group / N-lane / M-lane
    const int hi   = lane >> 4;            // half selector
    const int b0   = blockIdx.x * NB;      // first global batch of group

    // ---- stage te via async global->LDS (ASYNCcnt path, no VGPR bounce) ----
    {
        const float*   src  = task_embeds + (size_t)b0 * NT * DD;
        const uint32_t lds0 = (uint32_t)(uintptr_t)te_s;   // low 32b = LDS offset
        for (int i = lane; i < NB * NT * DD / 4; i += 32) {
            uint32_t ldsa = lds0 + (uint32_t)i * 16u;
            uint32_t goff = (uint32_t)i * 16u;
            asm volatile("global_load_async_to_lds_b128 %0, %1, %2"
                         :: "v"(ldsa), "v"(goff), "s"(src) : "memory");
        }
    }

    // ---- TDM demo: tensor_load_to_lds pulls 16 KB of W_upd into xT_s ----
    // (xT_s is fully rewritten in phase 2 before any read, so numerics are
    //  independent of this copy; it exercises the TENSORcnt path.)
    {
        const uint64_t ga   = (uint64_t)(uintptr_t)W_upd;
        const uint32_t ldsa = (uint32_t)(uintptr_t)xT_s;
        u32x4 g0 = { 1u,                                   // count = 1
                     ldsa,                                 // lds_addr
                     (uint32_t)ga,                         // global_addr[31:0]
                     (uint32_t)((ga >> 32) & 0x01FFFFFFu)  // global_addr[56:32]
                       | 0x80000000u };                    // type = 2 (image)
        i32x8 g1 = { (int)0x00020000,   // data_size = 4B
                     (int)0x10000000,   // tensor_dim0 = 4096 (bits 63:48)
                     (int)0x00010000,   // tensor_dim1 = 1    (bits 95:80)
                     (int)0x10000000,   // tile_dim0   = 4096 (bits 127:112)
                     (int)0x00000001,   // tile_dim1   = 1
                     (int)4096,         // tensor_dim0_stride low32
                     (int)0x10000000,   // tensor_dim1_stride low16 (bits 223:208)
                     0 };
        i32x4 gz = { 0, 0, 0, 0 };
#if defined(__clang_major__) && (__clang_major__ >= 23)
        i32x8 gz8 = { 0, 0, 0, 0, 0, 0, 0, 0 };
        __builtin_amdgcn_tensor_load_to_lds(g0, g1, gz, gz, gz8, 0);
#else
        __builtin_amdgcn_tensor_load_to_lds(g0, g1, gz, gz, 0);
#endif
    }

    for (int i = lane; i < NB * NT; i += 32) {
        nonag_s[i]  = nonag_g[(size_t)b0 * NT + i];
        counts_s[i] = 0.0f;
    }
    for (int i = lane; i < DD; i += 32) {
        wc0_s[i] = W_count[i];
        wc1_s[i] = W_count[DD + i];
        bc_s[i]  = b_count[i];
        bu_s[i]  = b_upd[i];
    }
    if (lane < NB) {
        uint32_t m = 0;
        for (int t = 0; t < NT; ++t)
            m |= (task_mask[(size_t)(b0 + lane) * NT + t] ? 1u : 0u) << t;
        mbits_s[lane] = m;
    }
    // warm W_upd (128 KB) into cache: one prefetch per 128B line
    for (int i = lane * 32; i < K2 * DD; i += 32 * 32)
        __builtin_prefetch(&W_upd[i], 0, 3);

    asm volatile("s_wait_asynccnt 0x0" ::: "memory");
    __builtin_amdgcn_s_wait_tensorcnt(0);
    __syncthreads();

    const float scale  = sqrtf((float)DD);
    const float NEGMIN = -3.402823466e38f;   // finfo(f32).min

    for (int s = 0; s < NA; ++s) {
        // ---- Phase 1: logits / softmax / argmax ----
        float ag[64];
        {
            const float* agp = agent_embeds + (((size_t)(b0 + b)) * NA + s) * DD + hi * 64;
            #pragma unroll
            for (int i = 0; i < 64; i += 4) {
                float4 v = *(const float4*)(agp + i);
                ag[i] = v.x; ag[i + 1] = v.y; ag[i + 2] = v.z; ag[i + 3] = v.w;
            }
        }
        // cnt_e is linear: ag.cnt_e = nonag*s0 + counts*s1 + s2
        float s0 = 0.f, s1 = 0.f, s2 = 0.f;
        #pragma unroll
        for (int i = 0; i < 64; ++i) {
            s0 = fmaf(ag[i], wc0_s[hi * 64 + i], s0);
            s1 = fmaf(ag[i], wc1_s[hi * 64 + i], s1);
            s2 = fmaf(ag[i], bc_s[hi * 64 + i], s2);
        }
        s0 += __shfl_xor(s0, 16);
        s1 += __shfl_xor(s1, 16);
        s2 += __shfl_xor(s2, 16);

        float z[NT];
        #pragma unroll
        for (int t = 0; t < NT; ++t) {
            const float4* tp = (const float4*)(te_s + (b * NT + t) * DD + hi * 64);
            float p = 0.f;
            #pragma unroll
            for (int i = 0; i < 16; ++i) {
                float4 v = tp[i];
                p = fmaf(ag[4 * i    ], v.x, p);
                p = fmaf(ag[4 * i + 1], v.y, p);
                p = fmaf(ag[4 * i + 2], v.z, p);
                p = fmaf(ag[4 * i + 3], v.w, p);
            }
            p += __shfl_xor(p, 16);
            z[t] = p;
        }

        if (lane < NB) {   // one lane finalizes each batch
            const uint32_t mb  = mbits_s[lane];
            const float*   gp  = gumbels + (((size_t)s) * B + (b0 + lane)) * NT;
            const float    agm = 1.0f - agent_mask[(size_t)(b0 + lane) * NA + s];
            float ex[NT];
            float zmax = -INFINITY;
            #pragma unroll
            for (int t = 0; t < NT; ++t) {
                float lg = fmaf(nonag_s[lane * NT + t], s0,
                            fmaf(counts_s[lane * NT + t], s1, z[t] + s2)) / scale;
                if ((mb >> t) & 1u) lg = NEGMIN;
                float v = lg + gp[t];
                z[t]  = v;
                zmax  = fmaxf(zmax, v);
            }
            float esum = 0.f;
            #pragma unroll
            for (int t = 0; t < NT; ++t) { ex[t] = expf(z[t] - zmax); esum += ex[t]; }
            int   tstar = 0;
            float smax  = -1.0f;
            #pragma unroll
            for (int t = 0; t < NT; ++t) {
                float st = ex[t] / esum;          // soft (matches jax exp/sum)
                ex[t] = st;
                if (st > smax) { smax = st; tstar = t; }   // first-tie argmax
            }
            float* op = out + (((size_t)(b0 + lane)) * NA + s) * NT;
            float hstar = 0.f;
            #pragma unroll
            for (int t = 0; t < NT; ++t) {
                float oh = (t == tstar) ? 1.0f : 0.0f;
                float h  = ((oh - ex[t]) + ex[t]) * agm;   // exact ref arithmetic
                op[t] = h;
                if (t == tstar) hstar = h;
            }
            counts_s[lane * NT + tstar] += hstar * 0.1f;
            ts_s[lane] = tstar;
            f_s[lane]  = hstar * agm;              // hstop * agm (ref applies agm twice)
        }
        __syncthreads();

        // ---- Phase 2: A rows  x[m] = relu(concat(te[m][t*], ag[m]))  (transposed) ----
        {
            const int    tb    = ts_s[b];
            const float* agrow = agent_embeds + (((size_t)(b0 + b)) * NA + s) * DD;
            const float* terow = te_s + (b * NT + tb) * DD;
            #pragma unroll 8
            for (int i = 0; i < 128; ++i) {
                int   k = 2 * i + hi;                      // lane-half parity stripes k
                float v = (k < DD) ? terow[k] : agrow[k - DD];
                xT_s[k * NB + b] = fmaxf(v, 0.0f);
            }
        }
        __syncthreads();

        // ---- Phase 3: WMMA  upd[16 x 128] = x(16x256) @ W_upd(256x128), f32 ----
        {
            const float* xa = xT_s + b + (2 * hi) * NB;            // A frag base
            const float* wb = W_upd + (size_t)(2 * hi) * DD + b;   // B frag base
            for (int nt = 0; nt < 8; ++nt) {
                v8f acc = {0.f, 0.f, 0.f, 0.f, 0.f, 0.f, 0.f, 0.f};
                const float* wbn = wb + nt * 16;
                #pragma unroll 8
                for (int ks = 0; ks < 64; ++ks) {
                    v2f a, bm;
                    a.x  = xa[(4 * ks    ) * NB];
                    a.y  = xa[(4 * ks + 1) * NB];
                    bm.x = wbn[(size_t)(4 * ks    ) * DD];
                    bm.y = wbn[(size_t)(4 * ks + 1) * DD];
                    acc  = __builtin_amdgcn_wmma_f32_16x16x4_f32(
                               false, a, false, bm, (short)0, acc, false, false);
                }
                // D layout: lane (n = b, half hi), VGPR j -> M = j + 8*hi
                const int   n   = nt * 16 + b;
                const float bun = bu_s[n];
                #pragma unroll
                for (int j = 0; j < 8; ++j) {
                    int   m  = j + 8 * hi;
                    float f  = f_s[m];
                    float* p = te_s + (m * NT + ts_s[m]) * DD + n;
                    *p = *p + (acc[j] + bun) * f;
                }
            }
        }
        __syncthreads();
    }
}

extern "C" void kernel_launch(void* const* d_in, const int* in_sizes, int n_in,
                              void* d_out, int out_size, void* d_ws, size_t ws_size,
                              hipStream_t stream) {
    const float*   task_embeds  = (const float*)  d_in[0];
    const float*   nonag        = (const float*)  d_in[1];
    const float*   agent_embeds = (const float*)  d_in[2];
    const uint8_t* task_mask    = (const uint8_t*)d_in[3];
    const float*   agent_mask   = (const float*)  d_in[4];
    const float*   gumbels      = (const float*)  d_in[5];
    const float*   W_count      = (const float*)  d_in[6];
    const float*   b_count      = (const float*)  d_in[7];
    const float*   W_upd        = (const float*)  d_in[8];
    const float*   b_upd        = (const float*)  d_in[9];
    float*         out          = (float*)        d_out;

    const int B = in_sizes[0] / (NT * DD);     // 4096
    dim3 grid(B / NB), block(32);
    hipLaunchKernelGGL(alloc_policy_kernel, grid, block, 0, stream,
                       task_embeds, nonag, agent_embeds, task_mask, agent_mask,
                       gumbels, W_count, b_count, W_upd, b_upd, out, B);
}